// WeightedMSE_4002909520771
// MI455X (gfx1250) — compile-verified
//
#include <hip/hip_runtime.h>

typedef float v2f __attribute__((ext_vector_type(2)));
typedef float v8f __attribute__((ext_vector_type(8)));
typedef float v4f __attribute__((ext_vector_type(4)));
typedef int   v4i __attribute__((ext_vector_type(4)));

#define WMSE_W 0.1f

// Per-element weighted squared error.
__device__ __forceinline__ float wmse_contrib(float p, int t) {
    float tf = (float)t;
    float d  = p - tf;
    bool mis = (p >= 0.5f) != (t == 1);
    float w  = mis ? (1.0f + WMSE_W) : 1.0f;
    return w * d * d;
}

// Reduce the 64 fp32 accumulators held by one wave (2 per lane) using a single
// V_WMMA_F32_16X16X4_F32 against an all-ones B matrix:
//   A (16x4, 2 VGPRs/lane) holds the 64 partials; D[m][n] = rowsum_m.
//   Summing the 8 D VGPRs in a lane covers rows 0-7 (lanes 0-15) or 8-15
//   (lanes 16-31); one shfl_xor(16) combines the halves -> full 64-way sum.
// Requires EXEC all-1s: called only from reconverged, uniform control flow.
__device__ __forceinline__ float wave_reduce_wmma(float acc0, float acc1) {
    v2f a; a.x = acc0; a.y = acc1;
    v2f b; b.x = 1.0f; b.y = 1.0f;
    v8f c = {};
    c = __builtin_amdgcn_wmma_f32_16x16x4_f32(
        /*neg_a=*/false, a, /*neg_b=*/false, b,
        /*c_mod=*/(short)0, c, /*reuse_a=*/false, /*reuse_b=*/false);
    float s = c[0] + c[1] + c[2] + c[3] + c[4] + c[5] + c[6] + c[7];
    s += __shfl_xor(s, 16, 32);
    return s;
}

// Pass 1: streaming grid-stride reduction over 128-bit vectors with
// non-temporal loads (268 MB stream > 192 MB L2 -> NT hint).
// One fp32 partial per block.
__global__ void __launch_bounds__(256)
wmse_partial_kernel(const v4f* __restrict__ pred4,
                    const v4i* __restrict__ tgt4,
                    float* __restrict__ partials, int n4) {
    float acc0 = 0.0f, acc1 = 0.0f;
    int stride = gridDim.x * blockDim.x;
    int gid    = blockIdx.x * blockDim.x + threadIdx.x;
    for (int i = gid; i < n4; i += stride) {
        v4f p = __builtin_nontemporal_load(&pred4[i]);
        v4i t = __builtin_nontemporal_load(&tgt4[i]);
        acc0 += wmse_contrib(p.x, t.x) + wmse_contrib(p.z, t.z);
        acc1 += wmse_contrib(p.y, t.y) + wmse_contrib(p.w, t.w);
    }
    // Wave-level 64-way reduction via WMMA (EXEC is full here).
    float s = wave_reduce_wmma(acc0, acc1);

    __shared__ float wave_sums[8];
    int wave = threadIdx.x >> 5;
    int lane = threadIdx.x & 31;
    if (lane == 0) wave_sums[wave] = s;
    __syncthreads();
    if (threadIdx.x == 0) {
        float bs = 0.0f;
        int nw = blockDim.x >> 5;
        for (int w = 0; w < nw; ++w) bs += wave_sums[w];
        partials[blockIdx.x] = bs;
    }
}

// Pass 2: single block folds the block partials (plus any scalar tail) and
// writes mean = sum / n. Also the only writer of d_out (self-initializing).
__global__ void __launch_bounds__(256)
wmse_final_kernel(const float* __restrict__ partials, int nb,
                  const float* __restrict__ pred,
                  const int*   __restrict__ tgt,
                  int tail_start, int n,
                  float* __restrict__ out, float inv_n) {
    float acc0 = 0.0f, acc1 = 0.0f;
    for (int i = threadIdx.x; i < nb; i += blockDim.x) {
        acc0 += partials[i];
    }
    // Scalar tail for n % 4 != 0 (none for the reference's 2^25, kept general).
    for (int i = tail_start + threadIdx.x; i < n; i += blockDim.x) {
        acc1 += wmse_contrib(pred[i], tgt[i]);
    }
    float s = wave_reduce_wmma(acc0, acc1);

    __shared__ float wave_sums[8];
    int wave = threadIdx.x >> 5;
    int lane = threadIdx.x & 31;
    if (lane == 0) wave_sums[wave] = s;
    __syncthreads();
    if (threadIdx.x == 0) {
        float total = 0.0f;
        int nw = blockDim.x >> 5;
        for (int w = 0; w < nw; ++w) total += wave_sums[w];
        out[0] = total * inv_n;
    }
}

extern "C" void kernel_launch(void* const* d_in, const int* in_sizes, int n_in,
                              void* d_out, int out_size, void* d_ws, size_t ws_size,
                              hipStream_t stream) {
    const float* pred = (const float*)d_in[0];
    const int*   tgt  = (const int*)d_in[1];
    float* out        = (float*)d_out;
    float* partials   = (float*)d_ws;

    int n  = in_sizes[0];
    int n4 = n >> 2;

    // 2048 blocks x 256 threads (8 wave32/block): each thread streams 16
    // float4+int4 pairs for N=2^25 -> enough in-flight loads to saturate HBM.
    int nb = 2048;
    size_t ws_floats = ws_size / sizeof(float);
    if ((size_t)nb > ws_floats) nb = (int)ws_floats;
    if (nb < 1) nb = 1;

    wmse_partial_kernel<<<nb, 256, 0, stream>>>(
        (const v4f*)pred, (const v4i*)tgt, partials, n4);

    wmse_final_kernel<<<1, 256, 0, stream>>>(
        partials, nb, pred, tgt, n4 << 2, n, out, 1.0f / (float)n);
}